// Mymodel_26242250179025
// MI455X (gfx1250) — compile-verified
//
#include <hip/hip_runtime.h>

#define NTOT  4900
#define NTOTP 4912              // 4900 padded to multiple of 16
#define NB1   708
#define NB2   4192
#define DIN   256
#define DOUTC 128

typedef float v2f __attribute__((ext_vector_type(2)));
typedef float v8f __attribute__((ext_vector_type(8)));

static __device__ __forceinline__ v8f wmma4(v2f a, v2f b, v8f c) {
  // D(16x16,f32) = A(16x4,f32) * B(4x16,f32) + C   -> v_wmma_f32_16x16x4_f32
  return __builtin_amdgcn_wmma_f32_16x16x4_f32(false, a, false, b, (short)0, c, false, false);
}

// ---------------------------------------------------------------------------
// Kernel A: fused projections  Hf/Q/K/V = feature @ W^T + b   (4900x256x128 x4)
// one wave = one 16x16 output tile, K-loop 256 in steps of 4 (WMMA f32 16x16x4)
// Edge rows handled by clamping the source row (pad rows never stored/consumed).
// ---------------------------------------------------------------------------
__global__ void proj_kernel(const float* __restrict__ feat,
    const float* __restrict__ Ww, const float* __restrict__ Wb,
    const float* __restrict__ Qw, const float* __restrict__ Qb,
    const float* __restrict__ Kw, const float* __restrict__ Kb,
    const float* __restrict__ Vw, const float* __restrict__ Vb,
    float* __restrict__ Hf, float* __restrict__ Q2d,
    float* __restrict__ K2d, float* __restrict__ V2d)
{
  const int lane = threadIdx.x & 31;
  const int wv   = threadIdx.x >> 5;
  const int ROWT = NTOTP / 16;                    // 307
  int tile = blockIdx.x * 8 + wv;
  if (tile >= ROWT * 32) return;
  int rt = tile >> 5;
  int ct = tile & 31;
  int wsel = ct >> 3;
  int o0   = (ct & 7) * 16;
  const float* W; const float* B; float* O;
  if      (wsel == 0) { W = Ww; B = Wb; O = Hf;  }
  else if (wsel == 1) { W = Qw; B = Qb; O = Q2d; }
  else if (wsel == 2) { W = Kw; B = Kb; O = K2d; }
  else                { W = Vw; B = Vb; O = V2d; }
  int lm = lane & 15;
  int hk = (lane >> 4) << 1;                      // 0 or 2
  int row = min(rt * 16 + lm, NTOT - 1);          // clamp: no divergence
  const float* arow = feat + (size_t)row * DIN + hk;
  const float* brow = W + (size_t)(o0 + lm) * DIN + hk;
  v8f acc; for (int v = 0; v < 8; v++) acc[v] = 0.f;
#pragma unroll
  for (int k = 0; k < DIN; k += 4) {
    v2f a = *(const v2f*)(arow + k);
    v2f b = *(const v2f*)(brow + k);
    acc = wmma4(a, b, acc);
  }
  float bb = B[o0 + lm];
  int mbase = rt * 16 + ((lane >> 4) << 3);
  for (int v = 0; v < 8; v++) {
    int gm = mbase + v;
    if (gm < NTOT) O[(size_t)gm * DOUTC + o0 + lm] = acc[v] + bb;
  }
}

// ---------------------------------------------------------------------------
// Kernel B: rr = r @ Wrw^T + Wrb   (4x128, tiny)
// ---------------------------------------------------------------------------
__global__ void rr_kernel(const float* __restrict__ r, const float* __restrict__ Wrw,
                          const float* __restrict__ Wrb, float* __restrict__ rr)
{
  int row = blockIdx.x, c = threadIdx.x;
  float s = Wrb[c];
  for (int k = 0; k < 256; k++) s += r[row * 256 + k] * Wrw[c * 256 + k];
  rr[row * 128 + c] = s;
}

// ---------------------------------------------------------------------------
// Kernel C1: blocked+scaled Q matrices.
// reshape map: blk(x,rel)[h,i,dh] = x[base + h*(n/4) + i/4, (i%4)*32+dh]*rel[col]
// QSa uses rr[1+rb], QSb uses rr[2+rb]; 0.5 folds in alpha/sqrt(H).
// ---------------------------------------------------------------------------
__global__ void qs_kernel(const float* __restrict__ Q2d, const float* __restrict__ rr,
                          float* __restrict__ QSa, float* __restrict__ QSb)
{
  int idx = blockIdx.x * 256 + threadIdx.x;
  if (idx >= NTOT * DOUTC) return;
  int i = idx >> 7, c = idx & 127;
  int h = c >> 5, dh = c & 31;
  int rb = (i < NB1) ? 0 : 1;
  int il = i - (rb ? NB1 : 0);
  int nq = rb ? (NB2 / 4) : (NB1 / 4);
  int base = rb ? NB1 : 0;
  int srow = base + h * nq + (il >> 2);
  int scol = ((il & 3) << 5) + dh;
  float qv = Q2d[(size_t)srow * DOUTC + scol] * 0.5f;
  QSa[idx] = qv * rr[(1 + rb) * 128 + scol];
  QSb[idx] = qv * rr[(2 + rb) * 128 + scol];
}

// ---------------------------------------------------------------------------
// Kernel C2: K^T blocked+scaled, stored transposed (128 x Ksz) for A-frags.
// top (k1, Ksz=708) scaled rr[1+rb]; bot (k2, Ksz=4192) scaled rr[2+rb].
// ---------------------------------------------------------------------------
__global__ void kt_kernel(const float* __restrict__ K2d, const float* __restrict__ rr,
                          float* __restrict__ KT0t, float* __restrict__ KT0b,
                          float* __restrict__ KT1t, float* __restrict__ KT1b)
{
  int idx = blockIdx.x * 256 + threadIdx.x;
  if (idx >= 2 * DOUTC * NTOT) return;
  int rb  = idx / (DOUTC * NTOT);
  int rem = idx % (DOUTC * NTOT);
  int c = rem / NTOT;
  int k = rem % NTOT;
  int h = c >> 5, dh = c & 31;
  if (k < NB1) {
    int srow = h * (NB1 / 4) + (k >> 2);
    int scol = ((k & 3) << 5) + dh;
    float v = K2d[(size_t)srow * DOUTC + scol] * rr[(1 + rb) * 128 + scol];
    (rb ? KT1t : KT0t)[(size_t)c * NB1 + k] = v;
  } else {
    int kl = k - NB1;
    int srow = NB1 + h * (NB2 / 4) + (kl >> 2);
    int scol = ((kl & 3) << 5) + dh;
    float v = K2d[(size_t)srow * DOUTC + scol] * rr[(2 + rb) * 128 + scol];
    (rb ? KT1b : KT0b)[(size_t)c * NB2 + kl] = v;
  }
}

// Kernel C3: hV blocked, padded to NTOTP rows (pad rows zero -> no guards later)
__global__ void vb_kernel(const float* __restrict__ V2d, float* __restrict__ VB)
{
  int idx = blockIdx.x * 256 + threadIdx.x;
  if (idx >= NTOTP * DOUTC) return;
  int j = idx >> 7, c = idx & 127;
  int h = c >> 5, dh = c & 31;
  VB[idx] = (j < NTOT)
      ? V2d[(size_t)(h * (NTOT / 4) + (j >> 2)) * DOUTC + ((j & 3) << 5) + dh]
      : 0.f;
}

// ---------------------------------------------------------------------------
// Kernel D: Z[sel] = KT[sel] @ adj_part   (M=128, K=708|4192, N=4900, padded 4912)
// sel = rb*2 + part. Block: 8 waves split K, each wave does all 8 M-tiles
// (B-frag of adj reused x8), fixed-order LDS reduction (deterministic).
// adj column clamped (pad columns of Z hold finite duplicates; harmless).
// ---------------------------------------------------------------------------
__global__ void zgemm_kernel(const float* __restrict__ adj,
    const float* __restrict__ KT0t, const float* __restrict__ KT0b,
    const float* __restrict__ KT1t, const float* __restrict__ KT1b,
    float* __restrict__ Z)
{
  __shared__ float red[2048];                       // 128 x 16 tile
  int sel = blockIdx.y;
  int rb = sel >> 1, part = sel & 1;
  const float* KT = part ? (rb ? KT1b : KT0b) : (rb ? KT1t : KT0t);
  int Ksz = part ? NB2 : NB1;
  int adjOff = part ? NB1 : 0;
  float* Zout = Z + (size_t)sel * (DOUTC * NTOTP);
  int j0 = blockIdx.x * 16;
  int lane = threadIdx.x & 31;
  int wv = threadIdx.x >> 5;
  int lm = lane & 15;
  int hk = (lane >> 4) << 1;
  int jcol = min(j0 + lm, NTOT - 1);                // clamp: no divergence
  int ksteps = Ksz >> 2;
  int chunk = (ksteps + 7) >> 3;
  int ks = wv * chunk;
  int ke = min(ksteps, ks + chunk);
  const float* bbase = adj + (size_t)adjOff * NTOT + jcol;
  v8f acc[8];
  for (int mt = 0; mt < 8; mt++) for (int v = 0; v < 8; v++) acc[mt][v] = 0.f;
  for (int kk = ks; kk < ke; kk++) {
    int k = (kk << 2) + hk;
    v2f b;
    b[0] = bbase[(size_t)k * NTOT];
    b[1] = bbase[(size_t)(k + 1) * NTOT];
#pragma unroll
    for (int mt = 0; mt < 8; mt++) {
      v2f a = *(const v2f*)(KT + (size_t)(mt * 16 + lm) * Ksz + k);
      acc[mt] = wmma4(a, b, acc[mt]);
    }
  }
  for (int t = threadIdx.x; t < 2048; t += 256) red[t] = 0.f;
  __syncthreads();
  int mb = (lane >> 4) << 3;
  for (int w = 0; w < 8; w++) {                     // fixed order -> deterministic
    if (wv == w)
      for (int mt = 0; mt < 8; mt++)
        for (int v = 0; v < 8; v++)
          red[(mt * 16 + mb + v) * 16 + lm] += acc[mt][v];
    __syncthreads();
  }
  for (int t = threadIdx.x; t < 2048; t += 256) {
    int m = t >> 4, n = t & 15;
    Zout[(size_t)m * NTOTP + j0 + n] = red[t];      // padded stride, no guard
  }
}

// ---------------------------------------------------------------------------
// Kernel E: fused alpha (K=64 per head) -> per-element softmax over 4 heads
// -> m accumulation (alpha_sm @ VB).  Row tiles split at rb boundary
// (45 tiles for rows<708, 262 for the rest). 8 waves split the 307 j'-tiles;
// alpha C-frags restaged through per-wave LDS into A-frag layout for m WMMAs.
// All loads unguarded: QS reads stay in-bounds (pad output rows discarded),
// Z has padded stride NTOTP, VB pad rows are zero (pad columns contribute 0).
// ---------------------------------------------------------------------------
__global__ void attn_kernel(const float* __restrict__ QSa, const float* __restrict__ QSb,
    const float* __restrict__ Z, const float* __restrict__ VB, float* __restrict__ mOut)
{
  __shared__ float aStage[8][256];
  __shared__ float mRed[2048];                      // 16 x 128
  int it = blockIdx.x;
  int rb, i0, valid;
  if (it < 45) { rb = 0; i0 = it * 16; valid = min(16, NB1 - i0); }
  else         { rb = 1; i0 = NB1 + (it - 45) * 16; valid = 16; }
  const float* Z0 = Z + (size_t)(rb * 2 + 0) * (DOUTC * NTOTP);
  const float* Z1 = Z + (size_t)(rb * 2 + 1) * (DOUTC * NTOTP);
  int lane = threadIdx.x & 31;
  int wv = threadIdx.x >> 5;
  int lm = lane & 15;
  int hk = (lane >> 4) << 1;
  // A row: stays < 4900 for every tile (tile 44 reads rb=1 rows whose alpha
  // output rows are discarded at store -> harmless, WMMA rows are independent)
  const float* qa = QSa + (size_t)(i0 + lm) * DOUTC;
  const float* qb = QSb + (size_t)(i0 + lm) * DOUTC;
  const int NJT = NTOTP / 16;                       // 307
  int chunk = (NJT + 7) >> 3;
  int jts = wv * chunk;
  int jte = min(NJT, jts + chunk);
  v8f macc[8];                                      // 4 heads x 2 dh-halves
  for (int f = 0; f < 8; f++) for (int v = 0; v < 8; v++) macc[f][v] = 0.f;

  for (int jt = jts; jt < jte; jt++) {
    int j0 = jt * 16;
    int jcol = j0 + lm;
    v8f a4[4];
    for (int h = 0; h < 4; h++) for (int v = 0; v < 8; v++) a4[h][v] = 0.f;
#pragma unroll
    for (int h = 0; h < 4; h++) {
#pragma unroll
      for (int cb = 0; cb < 2; cb++) {
        const float* Qs = cb ? qb : qa;
        const float* Zs = (cb ? Z1 : Z0) + jcol;
#pragma unroll
        for (int ksx = 0; ksx < 8; ksx++) {
          int kk = h * 32 + (ksx << 2) + hk;
          v2f aa = *(const v2f*)(Qs + kk);
          v2f bb;
          bb[0] = Zs[(size_t)kk * NTOTP];
          bb[1] = Zs[(size_t)(kk + 1) * NTOTP];
          a4[h] = wmma4(aa, bb, a4[h]);
        }
      }
    }
    // softmax over heads (elementwise across the 4 accumulators, same lane/VGPR slot)
    for (int v = 0; v < 8; v++) {
      float x0 = a4[0][v], x1 = a4[1][v], x2 = a4[2][v], x3 = a4[3][v];
      float mx = fmaxf(fmaxf(x0, x1), fmaxf(x2, x3));
      float e0 = __expf(x0 - mx), e1 = __expf(x1 - mx);
      float e2 = __expf(x2 - mx), e3 = __expf(x3 - mx);
      float inv = 1.0f / (e0 + e1 + e2 + e3);
      a4[0][v] = e0 * inv; a4[1][v] = e1 * inv; a4[2][v] = e2 * inv; a4[3][v] = e3 * inv;
    }
    int mb = (lane >> 4) << 3;
#pragma unroll
    for (int h = 0; h < 4; h++) {
      for (int v = 0; v < 8; v++) aStage[wv][(mb + v) * 16 + lm] = a4[h][v];
#pragma unroll
      for (int kx = 0; kx < 4; kx++) {
        v2f aa;
        aa[0] = aStage[wv][lm * 16 + (kx << 2) + hk];
        aa[1] = aStage[wv][lm * 16 + (kx << 2) + hk + 1];
        int jr = j0 + (kx << 2) + hk;                 // < 4912, VB pad rows zero
        const float* vrow = VB + (size_t)jr * DOUTC + h * 32 + lm;
#pragma unroll
        for (int t = 0; t < 2; t++) {
          v2f bb;
          bb[0] = vrow[t * 16];
          bb[1] = vrow[DOUTC + t * 16];
          macc[h * 2 + t] = wmma4(aa, bb, macc[h * 2 + t]);
        }
      }
    }
  }
  for (int t = threadIdx.x; t < 2048; t += 256) mRed[t] = 0.f;
  __syncthreads();
  int mb2 = (lane >> 4) << 3;
  for (int w = 0; w < 8; w++) {                     // fixed order -> deterministic
    if (wv == w)
      for (int h = 0; h < 4; h++)
        for (int t = 0; t < 2; t++)
          for (int v = 0; v < 8; v++)
            mRed[(mb2 + v) * 128 + h * 32 + t * 16 + lm] += macc[h * 2 + t][v];
    __syncthreads();
  }
  for (int t = threadIdx.x; t < 2048; t += 256) {
    int m = t >> 7, c = t & 127;
    if (m < valid) mOut[(size_t)(i0 + m) * DOUTC + c] = mRed[t];
  }
}

// ---------------------------------------------------------------------------
// Kernel F: per-row epilogue: relu->LN(1e-5) -> sigmoid gate -> LN(1e-6)->tanh -> m2w
// ---------------------------------------------------------------------------
__global__ void epilogue_kernel(const float* __restrict__ mIn, const float* __restrict__ Hf,
    const float* __restrict__ betaw, const float* __restrict__ betab,
    const float* __restrict__ ln_g, const float* __restrict__ ln_b,
    const float* __restrict__ m1w, const float* __restrict__ m1b,
    const float* __restrict__ mlng, const float* __restrict__ mlnb,
    const float* __restrict__ m2w, float* __restrict__ out)
{
  __shared__ float sA[128];
  __shared__ float sB[128];
  __shared__ float red[128];
  int row = blockIdx.x;
  int c = threadIdx.x;
  float mv = fmaxf(mIn[(size_t)row * 128 + c], 0.f);
  red[c] = mv; __syncthreads();
  for (int s = 64; s > 0; s >>= 1) { if (c < s) red[c] += red[c + s]; __syncthreads(); }
  float mu = red[0] * (1.f / 128.f); __syncthreads();
  float d = mv - mu;
  red[c] = d * d; __syncthreads();
  for (int s = 64; s > 0; s >>= 1) { if (c < s) red[c] += red[c + s]; __syncthreads(); }
  float var = red[0] * (1.f / 128.f); __syncthreads();
  float mhat = d * rsqrtf(var + 1e-5f) * ln_g[c] + ln_b[c];
  float hf = Hf[(size_t)row * 128 + c];
  sA[c] = hf; sB[c] = mhat; __syncthreads();
  float acc = betab[c];
  for (int k = 0; k < 128; k++) acc += sA[k] * betaw[c * 256 + k];
  for (int k = 0; k < 128; k++) acc += sB[k] * betaw[c * 256 + 128 + k];
  float beta = 1.f / (1.f + __expf(-acc));
  float outv = beta * mhat + (1.f - beta) * hf;
  __syncthreads();
  sA[c] = outv; __syncthreads();
  float t1 = m1b[c];
  for (int k = 0; k < 128; k++) t1 += sA[k] * m1w[c * 128 + k];
  red[c] = t1; __syncthreads();
  for (int s = 64; s > 0; s >>= 1) { if (c < s) red[c] += red[c + s]; __syncthreads(); }
  float mu2 = red[0] * (1.f / 128.f); __syncthreads();
  float d2 = t1 - mu2;
  red[c] = d2 * d2; __syncthreads();
  for (int s = 64; s > 0; s >>= 1) { if (c < s) red[c] += red[c + s]; __syncthreads(); }
  float var2 = red[0] * (1.f / 128.f); __syncthreads();
  float hv = tanhf(d2 * rsqrtf(var2 + 1e-6f) * mlng[c] + mlnb[c]);
  sB[c] = hv; __syncthreads();
  float fin = 0.f;
  for (int k = 0; k < 128; k++) fin += sB[k] * m2w[c * 128 + k];
  out[(size_t)row * 128 + c] = fin;
}

// ---------------------------------------------------------------------------
extern "C" void kernel_launch(void* const* d_in, const int* in_sizes, int n_in,
                              void* d_out, int out_size, void* d_ws, size_t ws_size,
                              hipStream_t stream)
{
  (void)in_sizes; (void)n_in; (void)out_size;
  const float* feature = (const float*)d_in[0];
  const float* adj   = (const float*)d_in[1];
  const float* r     = (const float*)d_in[2];
  const float* Ww    = (const float*)d_in[3];
  const float* Wb    = (const float*)d_in[4];
  const float* Wrw   = (const float*)d_in[5];
  const float* Wrb   = (const float*)d_in[6];
  const float* Qw    = (const float*)d_in[7];
  const float* Qb    = (const float*)d_in[8];
  const float* Kw    = (const float*)d_in[9];
  const float* Kb    = (const float*)d_in[10];
  const float* Vw    = (const float*)d_in[11];
  const float* Vb    = (const float*)d_in[12];
  const float* betaw = (const float*)d_in[13];
  const float* betab = (const float*)d_in[14];
  const float* ln_g  = (const float*)d_in[15];
  const float* ln_b  = (const float*)d_in[16];
  const float* m1w   = (const float*)d_in[17];
  const float* m1b   = (const float*)d_in[18];
  const float* mlng  = (const float*)d_in[19];
  const float* mlnb  = (const float*)d_in[20];
  const float* m2w   = (const float*)d_in[21];
  float* out = (float*)d_out;

  float* ws = (float*)d_ws;
  size_t off = 0;
  const size_t NM  = (size_t)NTOT * DOUTC;          // 627200
  const size_t NMP = (size_t)NTOTP * DOUTC;         // 628736
  float* Hf   = ws + off; off += NM;
  float* Q2d  = ws + off; off += NM;
  float* K2d  = ws + off; off += NM;
  float* V2d  = ws + off; off += NM;
  float* QSa  = ws + off; off += NM;
  float* QSb  = ws + off; off += NM;
  float* VBb  = ws + off; off += NMP;
  float* mB   = ws + off; off += NM;
  float* rrB  = ws + off; off += 512;
  float* KT0t = ws + off; off += (size_t)128 * NB1;
  float* KT0b = ws + off; off += (size_t)128 * NB2;
  float* KT1t = ws + off; off += (size_t)128 * NB1;
  float* KT1b = ws + off; off += (size_t)128 * NB2;
  float* Z    = ws + off; off += 4 * NMP;
  if (ws_size < off * sizeof(float)) return;        // ~35 MB required

  proj_kernel<<<dim3(1228), 256, 0, stream>>>(feature, Ww, Wb, Qw, Qb, Kw, Kb, Vw, Vb,
                                              Hf, Q2d, K2d, V2d);
  rr_kernel<<<4, 128, 0, stream>>>(r, Wrw, Wrb, rrB);
  qs_kernel<<<(NTOT * DOUTC + 255) / 256, 256, 0, stream>>>(Q2d, rrB, QSa, QSb);
  kt_kernel<<<(2 * DOUTC * NTOT + 255) / 256, 256, 0, stream>>>(K2d, rrB, KT0t, KT0b, KT1t, KT1b);
  vb_kernel<<<(NTOTP * DOUTC + 255) / 256, 256, 0, stream>>>(V2d, VBb);
  zgemm_kernel<<<dim3(307, 4), 256, 0, stream>>>(adj, KT0t, KT0b, KT1t, KT1b, Z);
  attn_kernel<<<307, 256, 0, stream>>>(QSa, QSb, Z, VBb, mB);
  epilogue_kernel<<<NTOT, 128, 0, stream>>>(mB, Hf, betaw, betab, ln_g, ln_b,
                                            m1w, m1b, mlng, mlnb, m2w, out);
}